// DifferentiableAggregation_avg_6330781794350
// MI455X (gfx1250) — compile-verified
//
#include <hip/hip_runtime.h>
#include <hip/hip_bf16.h>

// Reference constants
#define NSEG 262144          // NUM_SEGMENTS
#define KSHARP 10.0f         // sigmoid sharpness

// Native clang vector types (required by __builtin_nontemporal_load)
typedef float vf4 __attribute__((ext_vector_type(4)));
typedef int   vi4 __attribute__((ext_vector_type(4)));

// Workspace layout (all f32, zeroed each launch):
//   ws[0*NSEG + s] = sum of logits[:,0]             (s0)
//   ws[1*NSEG + s] = sum of logits[:,1]+logits[:,2] (s12)
//   ws[2*NSEG + s] = sum of row-max                 (rm)
//   ws[3*NSEG + s] = count                          (cnt)
//   ws[4*NSEG + s] = count of label==4              (c4)
//   ws[5*NSEG + s] = count of label==1              (c1)

__global__ __launch_bounds__(256) void scatter_k(
    const vf4* __restrict__ lg4,   // N/4 groups of 3 vf4 (12 floats = 4 rows)
    const vi4* __restrict__ idx4,
    const vi4* __restrict__ lab4,
    const vi4* __restrict__ fidx4,
    float* __restrict__ ws, int n4)
{
    float* __restrict__ s0  = ws;
    float* __restrict__ s12 = ws + 1 * NSEG;
    float* __restrict__ rm  = ws + 2 * NSEG;
    float* __restrict__ cnt = ws + 3 * NSEG;
    float* __restrict__ c4  = ws + 4 * NSEG;
    float* __restrict__ c1  = ws + 5 * NSEG;

    const int stride = gridDim.x * blockDim.x;
    for (int g = blockIdx.x * blockDim.x + threadIdx.x; g < n4; g += stride) {
        // Prefetch next grid-stride chunk of the logits stream (speculative;
        // OOB prefetches are silently dropped per ISA).
        __builtin_prefetch(&lg4[3 * (g + stride)], 0, 0);

        // Streamed inputs: non-temporal so they don't evict the L2-resident
        // 6.3MB accumulator table.
        vf4 a = __builtin_nontemporal_load(&lg4[3 * g + 0]);
        vf4 b = __builtin_nontemporal_load(&lg4[3 * g + 1]);
        vf4 c = __builtin_nontemporal_load(&lg4[3 * g + 2]);
        vi4 id = __builtin_nontemporal_load(&idx4[g]);
        vi4 lb = __builtin_nontemporal_load(&lab4[g]);
        vi4 fi = __builtin_nontemporal_load(&fidx4[g]);

        const float x[4] = { a.x, a.w, b.z, c.y };
        const float y[4] = { a.y, b.x, b.w, c.z };
        const float z[4] = { a.z, b.y, c.x, c.w };
        const int  ids[4] = { id.x, id.y, id.z, id.w };
        const int  lbs[4] = { lb.x, lb.y, lb.z, lb.w };
        const int  fis[4] = { fi.x, fi.y, fi.z, fi.w };

        #pragma unroll
        for (int r = 0; r < 4; ++r) {
            const int s = ids[r];
            const float mx = fmaxf(fmaxf(x[r], y[r]), z[r]);   // v_max3_f32
            // 4 no-return f32 atomics per row; table is L2-resident.
            unsafeAtomicAdd(&cnt[s], 1.0f);
            unsafeAtomicAdd(&rm[s],  mx);
            unsafeAtomicAdd(&s0[s],  x[r]);
            unsafeAtomicAdd(&s12[s], y[r] + z[r]);
            // Conditional label counts (expected ~0.4 atomics/row).
            const int l = lbs[r];
            if (l == 4)      unsafeAtomicAdd(&c4[fis[r]], 1.0f);
            else if (l == 1) unsafeAtomicAdd(&c1[fis[r]], 1.0f);
        }
    }
}

__global__ __launch_bounds__(256) void finalize_k(
    const float* __restrict__ ws, float2* __restrict__ out)
{
    const int s = blockIdx.x * blockDim.x + threadIdx.x;
    if (s >= NSEG) return;

    const float vs0  = ws[0 * NSEG + s];
    const float vs12 = ws[1 * NSEG + s];
    const float vrm  = ws[2 * NSEG + s];
    const float vcnt = ws[3 * NSEG + s];
    float vc4        = ws[4 * NSEG + s];
    float vc1        = ws[5 * NSEG + s];

    const float sc  = fmaxf(vcnt, 1.0f);
    const float avg = vrm / sc;
    const bool small = vcnt < 6.0f;
    vc4 = small ? vc4 : 0.0f;
    vc1 = small ? vc1 : 0.0f;

    const float a0 = KSHARP * (vs0  + vc1 * avg - 5.0f * avg);
    const float a1 = KSHARP * (vs12 + vc4 * avg - 1.0f * avg);
    const float j0 = 1.0f / (1.0f + __expf(-a0));   // v_exp_f32
    const float j1 = 1.0f / (1.0f + __expf(-a1));

    out[s] = make_float2(j0, j1);
}

extern "C" void kernel_launch(void* const* d_in, const int* in_sizes, int n_in,
                              void* d_out, int out_size, void* d_ws, size_t ws_size,
                              hipStream_t stream) {
    const float* logits = (const float*)d_in[0];   // [N,3] f32
    const int*   idx    = (const int*)d_in[1];     // [N] int
    const int*   lab    = (const int*)d_in[2];     // [N] int
    const int*   fidx   = (const int*)d_in[3];     // [N] int

    const int n  = in_sizes[1];    // N_SUB
    const int n4 = n >> 2;         // rows processed 4 at a time

    // Zero the 6.3MB accumulator table (graph-capturable memset node).
    (void)hipMemsetAsync(d_ws, 0, (size_t)6 * NSEG * sizeof(float), stream);

    // Grid-stride scatter: 4096 blocks x 256 threads (8 wave32 per block),
    // 4 iterations per thread over the 4M row-quads.
    scatter_k<<<4096, 256, 0, stream>>>(
        (const vf4*)logits, (const vi4*)idx, (const vi4*)lab,
        (const vi4*)fidx, (float*)d_ws, n4);

    finalize_k<<<(NSEG + 255) / 256, 256, 0, stream>>>(
        (const float*)d_ws, (float2*)d_out);
}